// LPMP_DeepSet_Expert_Core_50053548867787
// MI455X (gfx1250) — compile-verified
//
#include <hip/hip_runtime.h>

// ---------------------------------------------------------------------------
// LPMP DeepSet expert core, algebraically collapsed for MI455X (gfx1250).
//
//   q = floor(x*255/32) in {0..7}  (x ~ U[0,1), zero pad -> code 0)
//   out[p] = relu( Hist8(q of 3x3 nbhd)[p] @ U + b23 )
//   U  = (1/9) * relu(k*w1+b1)_{k=0..7} @ (w2 @ w3)   [8 x 64]
//   b23 = b2 @ w3 + b3                                [64]
//
// The [Npix,8]@[8,64] GEMM runs on V_WMMA_F32_16X16X4_F32 (two chained K=4
// steps, exact f32). Kernel is store-bandwidth bound (302 MB out @ 23.3 TB/s
// ~= 13 us floor), so stores are staged through LDS for full coalescing.
// ---------------------------------------------------------------------------

typedef float v2f __attribute__((ext_vector_type(2)));
typedef float v8f __attribute__((ext_vector_type(8)));

#define HDIM 384
#define WDIM 384
#define NBAT 8
#define QS   7.96875f   // 255/32
#define LPAD 132        // LDS row stride in floats (128+pad -> conflict-light)

// ---------------- Kernel 1: fold the MLP into U[8][64] and b23[64] ---------
__global__ __launch_bounds__(64) void lpmp_precompute_lut(
    const float* __restrict__ w1, const float* __restrict__ b1,
    const float* __restrict__ w2, const float* __restrict__ b2,
    const float* __restrict__ w3, const float* __restrict__ b3,
    float* __restrict__ ws) {
  const int n = threadIdx.x;                    // output channel 0..63
  // column n of W23 = w2 @ w3   (w2:[32,32] in->out, w3:[32,64] in->out)
  float w23[32];
  #pragma unroll
  for (int c = 0; c < 32; ++c) {
    float s = 0.f;
    for (int d = 0; d < 32; ++d) s += w2[c * 32 + d] * w3[d * 64 + n];
    w23[c] = s;
  }
  // b23 = b2 @ w3 + b3
  float bb = b3[n];
  for (int c = 0; c < 32; ++c) bb += b2[c] * w3[c * 64 + n];
  ws[512 + n] = bb;
  // U[k][n] = (1/9) * sum_c relu(k*w1[c]+b1[c]) * W23[c][n]
  for (int k = 0; k < 8; ++k) {
    float u = 0.f;
    for (int c = 0; c < 32; ++c) {
      float t = (float)k * w1[c] + b1[c];
      t = t > 0.f ? t : 0.f;
      u += t * w23[c];
    }
    ws[k * 64 + n] = u * (1.0f / 9.0f);
  }
}

// ---------------- Kernel 2: fused histogram + WMMA GEMM + coalesced store --
__global__ __launch_bounds__(256) void lpmp_fused_wmma(
    const float* __restrict__ x,   // [B,1,H,W]
    const float* __restrict__ ws,  // U[8][64] then b23[64]
    float* __restrict__ out) {     // [B,64,H,W]
  __shared__ float lds[64 * LPAD];             // [ch][pixel] tile, padded

  const int t    = threadIdx.x;
  const int lane = t & 31;
  const int wv   = t >> 5;                     // wave id 0..7 -> 16-pixel M tile
  const int w0   = blockIdx.x * 128;           // block covers 128 pixels of row h
  const int h    = blockIdx.y;
  const int b    = blockIdx.z;

  const int m  = lane & 15;                    // M index within 16x16 tile
  const int hi = lane >> 4;                    // 0: K pair {0,1}; 1: K pair {2,3}
  const int kb = hi * 2;                       // this lane's base K row
  const int px = w0 + wv * 16 + m;             // global w coordinate of pixel

  // --- per-pixel histogram of quantized 3x3 codes, only the 4 bins this
  //     lane owns in the A fragments: bins {kb, kb+1, kb+4, kb+5} ----------
  float c0 = 0.f, c1 = 0.f, c2 = 0.f, c3 = 0.f;
  const float* xb = x + (size_t)b * HDIM * WDIM;
  #pragma unroll
  for (int dh = -1; dh <= 1; ++dh) {
    #pragma unroll
    for (int dw = -1; dw <= 1; ++dw) {
      const int hh = h + dh, ww = px + dw;
      int q = 0;                               // zero pad -> code 0
      if (hh >= 0 && hh < HDIM && ww >= 0 && ww < WDIM) {
        float v = xb[hh * WDIM + ww];
        q = (int)(v * QS);                     // v >= 0 -> trunc == floor
        q = q > 7 ? 7 : q;
      }
      c0 += (float)(q == kb);
      c1 += (float)(q == kb + 1);
      c2 += (float)(q == kb + 4);
      c3 += (float)(q == kb + 5);
    }
  }
  // A fragments (16x4 f32 layout: lanes0-15 K=0/1 in v0/v1, lanes16-31 K=2/3)
  v2f aLo = {c0, c1};                          // K rows kb, kb+1
  v2f aHi = {c2, c3};                          // K rows kb+4, kb+5

  // --- 4 N-tiles of 16 channels: B = U[k][n]; C seeded with bias ----------
  const int nn = lane & 15;                    // N index within tile
  #pragma unroll
  for (int nt = 0; nt < 4; ++nt) {
    const int n = nt * 16 + nn;
    // B fragment layout assumed to mirror the documented A layout:
    // v0 = row kb (lanes0-15 K=0 / lanes16-31 K=2), v1 = row kb+1.
    v2f bLo = {ws[kb * 64 + n],       ws[(kb + 1) * 64 + n]};
    v2f bHi = {ws[(kb + 4) * 64 + n], ws[(kb + 5) * 64 + n]};
    const float bias = ws[512 + n];            // D[j] all share channel n
    v8f acc = {bias, bias, bias, bias, bias, bias, bias, bias};

    // D = A(K0..3) x B + D ; then A(K4..7) x B + D   (exact f32, K=8 total)
    acc = __builtin_amdgcn_wmma_f32_16x16x4_f32(false, aLo, false, bLo,
                                                (short)0, acc, false, false);
    acc = __builtin_amdgcn_wmma_f32_16x16x4_f32(false, aHi, false, bHi,
                                                (short)0, acc, false, false);

    // D layout: VGPR j -> pixel M=j (lanes0-15) / M=j+8 (lanes16-31), chan n
    const int prow = wv * 16 + hi * 8;
    #pragma unroll
    for (int j = 0; j < 8; ++j) {
      float v = acc[j];
      v = v > 0.f ? v : 0.f;                   // final relu
      lds[n * LPAD + prow + j] = v;
    }
  }
  __syncthreads();

  // --- coalesced channel-major store: each wave writes 512B-contiguous runs
  float* ob = out + ((size_t)b * 64) * (HDIM * WDIM) + h * WDIM + w0;
  #pragma unroll
  for (int r = 0; r < 8; ++r) {
    const int idx = r * 256 + t;               // 0..2047 float4s
    const int ch  = idx >> 5;                  // 32 float4 per channel row
    const int v4  = idx & 31;
    const float4 val = *(const float4*)&lds[ch * LPAD + v4 * 4];
    *(float4*)(ob + (size_t)ch * (HDIM * WDIM) + v4 * 4) = val;
  }
}

// ---------------------------------------------------------------------------
extern "C" void kernel_launch(void* const* d_in, const int* in_sizes, int n_in,
                              void* d_out, int out_size, void* d_ws, size_t ws_size,
                              hipStream_t stream) {
  const float* x  = (const float*)d_in[0];
  const float* w1 = (const float*)d_in[1];
  const float* b1 = (const float*)d_in[2];
  const float* w2 = (const float*)d_in[3];
  const float* b2 = (const float*)d_in[4];
  const float* w3 = (const float*)d_in[5];
  const float* b3 = (const float*)d_in[6];
  float* out = (float*)d_out;
  float* ws  = (float*)d_ws;                   // U[8][64] + b23[64] = 576 f32

  lpmp_precompute_lut<<<1, 64, 0, stream>>>(w1, b1, w2, b2, w3, b3, ws);

  dim3 grid(WDIM / 128, HDIM, NBAT);           // (3, 384, 8)
  lpmp_fused_wmma<<<grid, 256, 0, stream>>>(x, ws, out);
}